// HausdorffDistance_57999238365751
// MI455X (gfx1250) — compile-verified
//
#include <hip/hip_runtime.h>

typedef __attribute__((ext_vector_type(2))) float v2f;
typedef __attribute__((ext_vector_type(8))) float v8f;

#define BATCH 16
#define NPTS  4096
#define TILE  16
#define CTILES (NPTS / TILE)          // 256 column tiles
#define WAVES_PER_BLOCK 8
#define ROWS_PER_BLOCK (WAVES_PER_BLOCK * TILE)   // 128
#define ROWGROUPS (NPTS / ROWS_PER_BLOCK)          // 32

// ---------------------------------------------------------------------------
// Kernel 0: initialize workspace (colmin = +inf bits, rowmax = 0 bits).
// All stored distances are clamped >= 0, so uint ordering == float ordering.
// ---------------------------------------------------------------------------
__global__ void hd_init(unsigned* __restrict__ ws) {
    int i = blockIdx.x * blockDim.x + threadIdx.x;
    int total = BATCH * NPTS + BATCH;
    if (i < BATCH * NPTS) ws[i] = 0x7F800000u;   // +inf
    else if (i < total)   ws[i] = 0u;            // 0.0f
}

// ---------------------------------------------------------------------------
// Kernel 1: tiled pairwise sq-dist via V_WMMA_F32_16X16X4_F32.
// grid = (ROWGROUPS, BATCH), block = 256 threads (8 waves of 32).
// Each wave owns one 16-row pred tile and sweeps all 256 gt column tiles.
// ---------------------------------------------------------------------------
__global__ __launch_bounds__(256)
void hd_tiles(const float* __restrict__ preds,
              const float* __restrict__ gts,
              unsigned* __restrict__ colmin_g,   // [BATCH*NPTS] uint-bits of float
              unsigned* __restrict__ rowmax_g) { // [BATCH]
    __shared__ unsigned colmin_s[NPTS];          // 16 KB of the 320 KB WGP LDS

    const int bat  = blockIdx.y;
    const int tid  = threadIdx.x;
    const int wave = tid >> 5;
    const int lane = tid & 31;
    const int half = lane >> 4;                  // wave32: two 16-lane halves
    const int mrow = lane & 15;

    for (int i = tid; i < NPTS; i += blockDim.x) colmin_s[i] = 0x7F800000u;
    __syncthreads();

    // ---- A fragment: 16x4 f32 (K0=x, K1=y, K2=z, K3=0 pad) ----
    const int rowBase = (blockIdx.x * WAVES_PER_BLOCK + wave) * TILE;
    const float* pp = preds + ((size_t)bat * NPTS + rowBase + mrow) * 3;
    const float px = pp[0], py = pp[1], pz = pp[2];
    v2f a;
    a.x = half ? pz   : px;    // VGPR0: K=2 (hi half) / K=0 (lo half)
    a.y = half ? 0.0f : py;    // VGPR1: K=3 pad      / K=1
    const float rn = px * px + py * py + pz * pz;

    // rxv[r] = |pred[rowBase + 8*half + r]|^2  (lane M holds norm of row M)
    float rxv[8];
#pragma unroll
    for (int r = 0; r < 8; ++r)
        rxv[r] = __shfl(rn, half * 8 + r, 32);

    float rowmin[8];
#pragma unroll
    for (int r = 0; r < 8; ++r) rowmin[r] = __builtin_inff();

    const float* gbase = gts + (size_t)bat * NPTS * 3;

    for (int j = 0; j < CTILES; ++j) {
        // ---- B fragment: 4x16 f32, lane holds gt column N = mrow ----
        const float* gp = gbase + (size_t)(j * TILE + mrow) * 3;
        const float gx = gp[0], gy = gp[1], gz = gp[2];
        if (j + 1 < CTILES)
            __builtin_prefetch(gbase + (size_t)((j + 1) * TILE + mrow) * 3, 0, 3);
        v2f bf;
        bf.x = half ? gz   : gx;
        bf.y = half ? 0.0f : gy;
        const float ry = gx * gx + gy * gy + gz * gz;

        // z-tile: D[M=r+8*half, N=mrow] = sum_K A*B
        v8f c = {};
        c = __builtin_amdgcn_wmma_f32_16x16x4_f32(
                /*neg_a=*/false, a, /*neg_b=*/false, bf,
                /*c_mod=*/(short)0, c, /*reuse_a=*/false, /*reuse_b=*/false);

        float cm = __builtin_inff();
#pragma unroll
        for (int r = 0; r < 8; ++r) {
            float p = fmaxf(rxv[r] + ry - 2.0f * c[r], 0.0f); // sq-dist >= 0
            rowmin[r] = fminf(rowmin[r], p);
            cm = fminf(cm, p);
        }
        // Column min over M: lanes L and L+16 hold the two 8-row halves of the
        // same column N = mrow; both atomic-min the same LDS word, so the
        // ds_min_u32 itself performs the half-combine (no shuffle needed).
        atomicMin(&colmin_s[j * TILE + mrow], __float_as_uint(cm));
    }

    // ---- row min over N: butterfly across the 16 lanes of each half ----
#pragma unroll
    for (int r = 0; r < 8; ++r) {
        float v = rowmin[r];
        v = fminf(v, __shfl_xor(v, 1, 32));
        v = fminf(v, __shfl_xor(v, 2, 32));
        v = fminf(v, __shfl_xor(v, 4, 32));
        v = fminf(v, __shfl_xor(v, 8, 32));
        rowmin[r] = v;
    }
    float rmax = rowmin[0];
#pragma unroll
    for (int r = 1; r < 8; ++r) rmax = fmaxf(rmax, rowmin[r]);
    rmax = fmaxf(rmax, __shfl_xor(rmax, 16, 32));  // other half = other 8 rows
    if (lane == 0)
        atomicMax(&rowmax_g[bat], __float_as_uint(rmax));

    // ---- flush block-local column mins to global ----
    __syncthreads();
    for (int i = tid; i < NPTS; i += blockDim.x)
        atomicMin(&colmin_g[(size_t)bat * NPTS + i], colmin_s[i]);
}

// ---------------------------------------------------------------------------
// Kernel 2: per-batch max over column mins, combine with row max.
// ---------------------------------------------------------------------------
__global__ __launch_bounds__(256)
void hd_final(const unsigned* __restrict__ colmin_g,
              const unsigned* __restrict__ rowmax_g,
              float* __restrict__ out) {
    __shared__ float red[256];
    const int bat = blockIdx.x;
    const int tid = threadIdx.x;
    float m = 0.0f;
    for (int i = tid; i < NPTS; i += 256)
        m = fmaxf(m, __uint_as_float(colmin_g[(size_t)bat * NPTS + i]));
    red[tid] = m;
    __syncthreads();
    for (int s = 128; s > 0; s >>= 1) {
        if (tid < s) red[tid] = fmaxf(red[tid], red[tid + s]);
        __syncthreads();
    }
    if (tid == 0)
        out[bat] = 0.5f * (red[0] + __uint_as_float(rowmax_g[bat]));
}

extern "C" void kernel_launch(void* const* d_in, const int* in_sizes, int n_in,
                              void* d_out, int out_size, void* d_ws, size_t ws_size,
                              hipStream_t stream) {
    const float* preds = (const float*)d_in[0];
    const float* gts   = (const float*)d_in[1];
    float* out = (float*)d_out;

    unsigned* colmin = (unsigned*)d_ws;               // BATCH*NPTS uints
    unsigned* rowmax = colmin + (size_t)BATCH * NPTS; // BATCH uints

    const int initN = BATCH * NPTS + BATCH;
    hd_init<<<(initN + 255) / 256, 256, 0, stream>>>(colmin);
    hd_tiles<<<dim3(ROWGROUPS, BATCH), 256, 0, stream>>>(preds, gts, colmin, rowmax);
    hd_final<<<BATCH, 256, 0, stream>>>(colmin, rowmax, out);
}